// GraphConvNet_30124900614317
// MI455X (gfx1250) — compile-verified
//
#include <hip/hip_runtime.h>

// ---------------------------------------------------------------------------
// GraphConvNet forward for MI455X (gfx1250, wave32).
//  - Scatter/gather kernels: edge-parallel, coalesced, rely on 192MB L2 to
//    keep the 25.6MB node arrays resident (HBM only streams the edge list).
//  - Node GEMMs (H=64): V_WMMA_F32_16X16X4_F32, one 16x16 tile per wave,
//    K=64 -> 16 wmma steps per weight matrix; weights staged in LDS.
//  - BN statistics fused into GEMMs via LDS ds_add_f32 reduction.
// ---------------------------------------------------------------------------

namespace {
constexpr int   Nn   = 100000;   // nodes
constexpr int   Ee   = 1600000;  // edges
constexpr int   Fin  = 9;        // input features
constexpr int   H    = 64;       // hidden
constexpr int   Oc   = 6;        // output classes
constexpr int   Gg   = 1024;     // graphs
constexpr float EPSf = 1e-5f;
}

typedef float v2f __attribute__((ext_vector_type(2)));
typedef float v8f __attribute__((ext_vector_type(8)));

__global__ void k_zero(float* __restrict__ p, int n) {
  int i = blockIdx.x * blockDim.x + threadIdx.x;
  if (i < n) p[i] = 0.0f;
}

// Layer-0 scatter: one thread per edge, 9 feature atomics (x is 3.6MB, L2-hot).
__global__ void k_scatter9(const float* __restrict__ x, const int* __restrict__ ei,
                           const float* __restrict__ ew, float* __restrict__ agg) {
  int e = blockIdx.x * blockDim.x + threadIdx.x;
  if (e >= Ee) return;
  int s = ei[e];
  int d = ei[Ee + e];
  float w = ew[e];
  const float* xs = x + (size_t)s * Fin;
  float* ad = agg + (size_t)d * Fin;
#pragma unroll
  for (int k = 0; k < Fin; ++k) atomicAdd(&ad[k], xs[k] * w);
}

// 64-wide scatter: thread = (edge, feature). Lanes of a wave read contiguous
// 128B of the source row; atomic adds land in L2 (node array is L2-resident).
__global__ void k_scatter64(const float* __restrict__ h, const int* __restrict__ ei,
                            const float* __restrict__ ew, float* __restrict__ agg) {
  int idx = blockIdx.x * blockDim.x + threadIdx.x;   // < E*64 = 102.4M
  int e = idx >> 6;
  int f = idx & 63;
  int s = ei[e];
  int d = ei[Ee + e];
  float w = ew[e];
  atomicAdd(&agg[(size_t)d * H + f], h[(size_t)s * H + f] * w);
}

// Layer-0 GEMM (K=9, scalar is fine) + fused BN statistics.
__global__ void k_gemm0_stats(const float* __restrict__ agg, const float* __restrict__ x,
                              const float* __restrict__ Wrel, const float* __restrict__ Wroot,
                              const float* __restrict__ bias, float* __restrict__ pre,
                              float* __restrict__ gSum, float* __restrict__ gSq) {
  __shared__ float sSum[H], sSq[H];
  int tid = threadIdx.x;
  if (tid < H) { sSum[tid] = 0.0f; sSq[tid] = 0.0f; }
  __syncthreads();
  int idx = blockIdx.x * blockDim.x + tid;           // N*H exact
  int n = idx >> 6, f = idx & 63;
  float acc = bias[f];
  const float* an = agg + (size_t)n * Fin;
  const float* xn = x   + (size_t)n * Fin;
#pragma unroll
  for (int k = 0; k < Fin; ++k)
    acc = fmaf(an[k], Wrel[k * H + f], fmaf(xn[k], Wroot[k * H + f], acc));
  pre[(size_t)n * H + f] = acc;
  atomicAdd(&sSum[f], acc);
  atomicAdd(&sSq[f], acc * acc);
  __syncthreads();
  if (tid < H) { atomicAdd(&gSum[tid], sSum[tid]); atomicAdd(&gSq[tid], sSq[tid]); }
}

// Layers 1/2 GEMM:  pre = agg @ Wrel + hin @ Wroot + bias  via WMMA f32.
// Block = 256 threads = 8 waves; each wave owns one 16x16 output tile
// (2 row-tiles x 4 col-tiles => 32 node rows per block). Weights in LDS.
// Fragment layout per CDNA5 ISA 7.12.2 (32-bit A 16x4 / B 4x16 / C 16x16).
__global__ void k_gemm_wmma(const float* __restrict__ agg, const float* __restrict__ hin,
                            const float* __restrict__ Wrel, const float* __restrict__ Wroot,
                            const float* __restrict__ bias, float* __restrict__ pre,
                            float* __restrict__ gSum, float* __restrict__ gSq, int doStats) {
  __shared__ float sWrel[H * H];
  __shared__ float sWroot[H * H];
  __shared__ float sSum[H], sSq[H];
  int tid = threadIdx.x;
  for (int i = tid; i < H * H; i += 256) { sWrel[i] = Wrel[i]; sWroot[i] = Wroot[i]; }
  if (tid < H) { sSum[tid] = 0.0f; sSq[tid] = 0.0f; }
  __syncthreads();

  int wave = tid >> 5, lane = tid & 31;
  int n0 = blockIdx.x * 32 + ((wave >> 2) << 4);   // node-row tile base
  int c0 = (wave & 3) << 4;                        // feature-col tile base
  int m  = lane & 15;                              // A row / B,C,D column in tile
  int kh = (lane >> 4) << 1;                       // lane-half K offset: 0 or 2
  int col = c0 + m;

  const float* aRel  = agg + (size_t)(n0 + m) * H;
  const float* aRoot = hin + (size_t)(n0 + m) * H;
  const float* bRel  = sWrel  + col;
  const float* bRoot = sWroot + col;

  float bv = bias[col];                            // bias depends on column only
  v8f acc;
#pragma unroll
  for (int r = 0; r < 8; ++r) acc[r] = bv;

#pragma unroll
  for (int k = 0; k < H; k += 4) {
    v2f a0 = { aRel[k + kh], aRel[k + kh + 1] };
    v2f b0 = { bRel[(size_t)(k + kh) * H], bRel[(size_t)(k + kh + 1) * H] };
    acc = __builtin_amdgcn_wmma_f32_16x16x4_f32(false, a0, false, b0, (short)0, acc, false, false);
    v2f a1 = { aRoot[k + kh], aRoot[k + kh + 1] };
    v2f b1 = { bRoot[(size_t)(k + kh) * H], bRoot[(size_t)(k + kh + 1) * H] };
    acc = __builtin_amdgcn_wmma_f32_16x16x4_f32(false, a1, false, b1, (short)0, acc, false, false);
  }

  int rbase = n0 + ((lane >> 4) << 3);             // C/D: lanes>=16 hold M=r+8
#pragma unroll
  for (int r = 0; r < 8; ++r) {
    float v = acc[r];
    pre[(size_t)(rbase + r) * H + col] = v;
    if (doStats) { atomicAdd(&sSum[col], v); atomicAdd(&sSq[col], v * v); }
  }
  __syncthreads();
  if (doStats && tid < H) { atomicAdd(&gSum[tid], sSum[tid]); atomicAdd(&gSq[tid], sSq[tid]); }
}

// BatchNorm (from fused sums) + ReLU.
__global__ void k_bn_relu(const float* __restrict__ pre, const float* __restrict__ gSum,
                          const float* __restrict__ gSq, const float* __restrict__ gamma,
                          const float* __restrict__ beta, float* __restrict__ hout) {
  int idx = blockIdx.x * blockDim.x + threadIdx.x;  // N*H exact
  int f = idx & 63;
  constexpr float invN = 1.0f / (float)Nn;
  float mean = gSum[f] * invN;
  float var  = gSq[f] * invN - mean * mean;
  float sc = rsqrtf(var + EPSf) * gamma[f];
  float sh = beta[f] - mean * sc;
  hout[idx] = fmaxf(fmaf(pre[idx], sc, sh), 0.0f);
}

// Global mean-pool accumulation (batch is sorted; atomics hit L2-resident 256KB).
__global__ void k_pool(const float* __restrict__ h3, const int* __restrict__ batch,
                       float* __restrict__ pooled, float* __restrict__ counts) {
  int idx = blockIdx.x * blockDim.x + threadIdx.x;  // N*H exact
  int n = idx >> 6, f = idx & 63;
  int g = batch[n];
  atomicAdd(&pooled[(size_t)g * H + f], h3[idx]);
  if (f == 0) atomicAdd(&counts[g], 1.0f);
}

// MLP head: one 64-thread block per graph.
__global__ void k_head(const float* __restrict__ pooled, const float* __restrict__ counts,
                       const float* __restrict__ Wl1, const float* __restrict__ bl1,
                       const float* __restrict__ Wl2, const float* __restrict__ bl2,
                       float* __restrict__ out) {
  __shared__ float row[H];
  __shared__ float z[H];
  int g = blockIdx.x, f = threadIdx.x;
  float cnt = fmaxf(counts[g], 1.0f);
  row[f] = pooled[(size_t)g * H + f] / cnt;
  __syncthreads();
  float s = bl1[f];
#pragma unroll 8
  for (int k = 0; k < H; ++k) s = fmaf(row[k], Wl1[k * H + f], s);
  z[f] = fmaxf(s, 0.0f);
  __syncthreads();
  if (f < Oc) {
    float o = bl2[f];
#pragma unroll 8
    for (int k = 0; k < H; ++k) o = fmaf(z[k], Wl2[k * Oc + f], o);
    out[g * Oc + f] = o;
  }
}

extern "C" void kernel_launch(void* const* d_in, const int* in_sizes, int n_in,
                              void* d_out, int out_size, void* d_ws, size_t ws_size,
                              hipStream_t stream) {
  (void)in_sizes; (void)n_in; (void)out_size; (void)ws_size;
  const float* x      = (const float*)d_in[0];
  const int*   ei     = (const int*)d_in[1];
  const float* ew     = (const float*)d_in[2];
  const int*   batch  = (const int*)d_in[3];
  const float* Wrel0  = (const float*)d_in[4];
  const float* Wroot0 = (const float*)d_in[5];
  const float* b0     = (const float*)d_in[6];
  const float* Wrel1  = (const float*)d_in[7];
  const float* Wroot1 = (const float*)d_in[8];
  const float* b1     = (const float*)d_in[9];
  const float* Wrel2  = (const float*)d_in[10];
  const float* Wroot2 = (const float*)d_in[11];
  const float* b2     = (const float*)d_in[12];
  const float* g0     = (const float*)d_in[13];
  const float* be0    = (const float*)d_in[14];
  const float* g1     = (const float*)d_in[15];
  const float* be1    = (const float*)d_in[16];
  const float* Wl1    = (const float*)d_in[17];
  const float* bl1    = (const float*)d_in[18];
  const float* Wl2    = (const float*)d_in[19];
  const float* bl2    = (const float*)d_in[20];
  float* out = (float*)d_out;

  // Workspace layout (floats): 3 rotating N*H buffers + stats + pooled + counts.
  size_t NH = (size_t)Nn * H;                        // 6.4M floats
  float* buf0   = (float*)d_ws;                      // agg
  float* buf1   = buf0 + NH;                         // activations h
  float* buf2   = buf1 + NH;                         // pre-activation
  float* stats  = buf2 + NH;                         // 4 x 64 (sum0,sq0,sum1,sq1)
  float* pooled = stats + 256;                       // G*H
  float* counts = pooled + (size_t)Gg * H;           // G

  dim3 b256(256);
  int nhBlocks   = (int)(NH / 256);                  // 25000
  int edgeBlocks = (int)(((size_t)Ee * H) / 256);    // 400000
  int smallN     = 256 + Gg * H + Gg;

  // ---- Layer 0: scatter(x) -> GEMM(K=9)+stats -> BN+ReLU ----
  k_zero<<<(Nn * Fin + 255) / 256, b256, 0, stream>>>(buf0, Nn * Fin);
  k_zero<<<(smallN + 255) / 256, b256, 0, stream>>>(stats, smallN);
  k_scatter9<<<(Ee + 255) / 256, b256, 0, stream>>>(x, ei, ew, buf0);
  k_gemm0_stats<<<nhBlocks, b256, 0, stream>>>(buf0, x, Wrel0, Wroot0, b0, buf2,
                                               stats, stats + 64);
  k_bn_relu<<<nhBlocks, b256, 0, stream>>>(buf2, stats, stats + 64, g0, be0, buf1);

  // ---- Layer 1: scatter -> WMMA GEMM+stats -> BN+ReLU ----
  k_zero<<<nhBlocks, b256, 0, stream>>>(buf0, (int)NH);
  k_scatter64<<<edgeBlocks, b256, 0, stream>>>(buf1, ei, ew, buf0);
  k_gemm_wmma<<<Nn / 32, b256, 0, stream>>>(buf0, buf1, Wrel1, Wroot1, b1, buf2,
                                            stats + 128, stats + 192, 1);
  k_bn_relu<<<nhBlocks, b256, 0, stream>>>(buf2, stats + 128, stats + 192, g1, be1, buf1);

  // ---- Layer 2: scatter -> WMMA GEMM (no norm) ----
  k_zero<<<nhBlocks, b256, 0, stream>>>(buf0, (int)NH);
  k_scatter64<<<edgeBlocks, b256, 0, stream>>>(buf1, ei, ew, buf0);
  k_gemm_wmma<<<Nn / 32, b256, 0, stream>>>(buf0, buf1, Wrel2, Wroot2, b2, buf2,
                                            stats + 128, stats + 192, 0);

  // ---- Pool + head ----
  k_pool<<<nhBlocks, b256, 0, stream>>>(buf2, batch, pooled, counts);
  k_head<<<Gg, dim3(64), 0, stream>>>(pooled, counts, Wl1, bl1, Wl2, bl2, out);
}